// Quantizer_GPTQ_42288247996921
// MI455X (gfx1250) — compile-verified
//
#include <hip/hip_runtime.h>

// GPTQ 4-bit quantizer grid search, one row per block, LDS-resident row.
// CDNA5 paths: async global->LDS staging (ASYNCcnt), wave32 shuffle reductions,
// b128 LDS reads in the transcendental-bound hot loop.

#define COLS        11008
#define COLS4       (COLS / 4)           // 2752 float4
#define BLOCK       256
#define WAVESZ      32
#define NWAVE       (BLOCK / WAVESZ)     // 8
#define NSTEPS      80                   // int(0.8 * 100)
#define PER_THREAD  (COLS / BLOCK)       // 43 (exact)

#if defined(__HIP_DEVICE_COMPILE__) && \
    __has_builtin(__builtin_amdgcn_global_load_async_to_lds_b32) && \
    __has_builtin(__builtin_amdgcn_s_wait_asynccnt)
#define USE_ASYNC_LDS 1
#else
#define USE_ASYNC_LDS 0
#endif

#if defined(__HIP_DEVICE_COMPILE__) && \
    __has_builtin(__builtin_amdgcn_exp2f) && __has_builtin(__builtin_amdgcn_logf)
#define FAST_EXP2(v) __builtin_amdgcn_exp2f(v)   // v_exp_f32
#define FAST_LOG2(v) __builtin_amdgcn_logf(v)    // v_log_f32 (base-2)
#else
#define FAST_EXP2(v) exp2f(v)
#define FAST_LOG2(v) log2f(v)
#endif

// |quantize-dequantize error|^2.4 for one element.
__device__ __forceinline__ float qerr_term(float v, float inv, float scale1,
                                           float zero, float maxq) {
    float q = __builtin_rintf(v * inv) + zero;       // round-half-even == jnp.round
    q = fminf(fmaxf(q, 0.0f), maxq);
    const float d  = scale1 * (q - zero) - v;
    // |d|^2.4 = exp2(2.4*log2|d|); d==0 -> log2->-inf -> exp2->0 (exact)
    return FAST_EXP2(2.4f * FAST_LOG2(fabsf(d)));
}

__device__ __forceinline__ float deq_term(float v, float bscale, float zero, float maxq) {
    float q = __builtin_rintf(v / bscale) + zero;
    q = fminf(fmaxf(q, 0.0f), maxq);
    return bscale * (q - zero);
}

__global__ __launch_bounds__(BLOCK) void gptq_quant_kernel(
    const float* __restrict__ x, const int* __restrict__ bits_p,
    float* __restrict__ out)
{
    __shared__ __align__(16) float s_row[COLS];
    __shared__ float s_err[NSTEPS];
    __shared__ float s_mn[NWAVE];
    __shared__ float s_mx[NWAVE];

    const int t    = threadIdx.x;
    const int lane = t & (WAVESZ - 1);
    const int wave = t >> 5;
    const long long base = (long long)blockIdx.x * COLS;

    const float maxq = (float)((1 << *bits_p) - 1);   // 15.0 for bits=4
    const float zero = (maxq + 1.0f) * 0.5f;          // 8.0

    // ---- Phase 1: stage the row into LDS (HBM touched exactly once) ----
#if USE_ASYNC_LDS
    {
        const float* gp = x + base;
        #pragma unroll
        for (int k = 0; k < PER_THREAD; ++k) {
            const int idx = t + BLOCK * k;
            __builtin_amdgcn_global_load_async_to_lds_b32(
                (__attribute__((address_space(1))) int*)(gp + idx),
                (__attribute__((address_space(3))) int*)(&s_row[idx]),
                0, 0);
        }
        __builtin_amdgcn_s_wait_asynccnt(0);
    }
#else
    #pragma unroll 4
    for (int k = 0; k < PER_THREAD; ++k) {
        const int idx = t + BLOCK * k;
        s_row[idx] = x[base + idx];
    }
#endif
    __syncthreads();

    const float4* row4 = reinterpret_cast<const float4*>(s_row);

    // ---- Phase 2: block-wide row min/max (b128 LDS reads) ----
    float mn = INFINITY, mx = -INFINITY;
    for (int c = t; c < COLS4; c += BLOCK) {           // 10-11 iters/thread
        const float4 v = row4[c];
        mn = fminf(mn, fminf(fminf(v.x, v.y), fminf(v.z, v.w)));
        mx = fmaxf(mx, fmaxf(fmaxf(v.x, v.y), fmaxf(v.z, v.w)));
    }
    #pragma unroll
    for (int off = WAVESZ / 2; off > 0; off >>= 1) {
        mn = fminf(mn, __shfl_xor(mn, off, WAVESZ));
        mx = fmaxf(mx, __shfl_xor(mx, off, WAVESZ));
    }
    if (lane == 0) { s_mn[wave] = mn; s_mx[wave] = mx; }
    __syncthreads();
    mn = s_mn[0]; mx = s_mx[0];
    #pragma unroll
    for (int w = 1; w < NWAVE; ++w) {
        mn = fminf(mn, s_mn[w]);
        mx = fmaxf(mx, s_mx[w]);
    }

    // Symmetrize exactly like the reference (order matters).
    float xmin = fminf(mn, 0.0f);
    float xmax = fmaxf(mx, 0.0f);
    xmax = fmaxf(fabsf(xmin), xmax);
    xmin = (xmin < 0.0f) ? -xmax : xmin;
    if (xmin == 0.0f && xmax == 0.0f) { xmin = -1.0f; xmax = 1.0f; }

    // ---- Phase 3: 80-step shrink search; each wave owns 10 independent steps ----
    for (int s = wave; s < NSTEPS; s += NWAVE) {
        const float p      = 1.0f - (float)s / 100.0f;
        const float scale1 = (p * xmax - p * xmin) / maxq;   // > 0 always (p >= 0.21)
        const float inv    = 1.0f / scale1;
        float err = 0.0f;
        #pragma unroll 2
        for (int c = lane; c < COLS4; c += WAVESZ) {         // 86 iters, ds_load_b128
            const float4 v = row4[c];
            err += qerr_term(v.x, inv, scale1, zero, maxq);
            err += qerr_term(v.y, inv, scale1, zero, maxq);
            err += qerr_term(v.z, inv, scale1, zero, maxq);
            err += qerr_term(v.w, inv, scale1, zero, maxq);
        }
        #pragma unroll
        for (int off = WAVESZ / 2; off > 0; off >>= 1)
            err += __shfl_xor(err, off, WAVESZ);
        if (lane == 0) s_err[s] = err;
    }
    __syncthreads();

    // ---- Phase 4: sequential argmin (replicates reference strict-< update order) ----
    float best   = INFINITY;
    float bscale = (xmax - xmin) / maxq;                     // scale0
    for (int s = 0; s < NSTEPS; ++s) {
        const float p  = 1.0f - (float)s / 100.0f;
        const float sc = (p * xmax - p * xmin) / maxq;
        const float e  = s_err[s];
        if (e < best) { best = e; bscale = sc; }
    }

    // ---- Phase 5: final quantize-dequantize, b128 stores ----
    float4* out4 = reinterpret_cast<float4*>(out) + (long long)blockIdx.x * COLS4;
    for (int c = t; c < COLS4; c += BLOCK) {
        const float4 v = row4[c];
        float4 r;
        r.x = deq_term(v.x, bscale, zero, maxq);
        r.y = deq_term(v.y, bscale, zero, maxq);
        r.z = deq_term(v.z, bscale, zero, maxq);
        r.w = deq_term(v.w, bscale, zero, maxq);
        out4[c] = r;
    }
}

extern "C" void kernel_launch(void* const* d_in, const int* in_sizes, int n_in,
                              void* d_out, int out_size, void* d_ws, size_t ws_size,
                              hipStream_t stream) {
    const float* x    = (const float*)d_in[0];
    const int*   bits = (const int*)d_in[1];
    float*       out  = (float*)d_out;

    const int rows = in_sizes[0] / COLS;   // 4096
    gptq_quant_kernel<<<dim3(rows), dim3(BLOCK), 0, stream>>>(x, bits, out);
}